// VoxelQuerySampler_13907104104882
// MI455X (gfx1250) — compile-verified
//
#include <hip/hip_runtime.h>
#include <cstdint>

// ---------------- problem geometry (from reference) ----------------
#define VX_NX 200
#define VX_NY 200
#define VX_NZ 16
#define NPTS  (VX_NZ * VX_NX * VX_NY)   // 640000
#define IMG_H 32
#define IMG_W 88
#define HW    (IMG_H * IMG_W)           // 2816 texels per channel
#define NCH   64

// ---------------- kernel tiling ----------------
#define GC    4                          // channels staged in LDS per step (45 KB)
#define NSTEP (NCH / GC)                 // 16
#define TPB   256                        // 8 waves (wave32)
#define PPT   4                          // points per thread
#define PTS_PER_BLOCK (TPB * PPT)        // 1024 consecutive voxel indices

typedef __attribute__((ext_vector_type(4))) int i32x4;
typedef __attribute__((ext_vector_type(8))) int i32x8;

// 2-D TDM load: `rows` lines of `elems` fp32 each, line stride `row_stride`
// elements in global memory, packed contiguously into LDS at lds_byte.
// Descriptor layout per CDNA5 ISA ch.8 (D# groups 0/1; 2-D tensor -> groups 2/3 unused).
__device__ __forceinline__ void tdm_load_rows_to_lds(unsigned lds_byte,
                                                     const float* src,
                                                     unsigned elems,
                                                     unsigned rows,
                                                     unsigned row_stride)
{
    unsigned long long ga = (unsigned long long)(uintptr_t)src;
    i32x4 g0 = {
        (int)1,                                                   // count=1 (valid user D#)
        (int)lds_byte,                                            // lds_addr [63:32]
        (int)(unsigned)(ga & 0xFFFFFFFFull),                      // global_addr[31:0]
        (int)((unsigned)((ga >> 32) & 0x1FFFFFFull) | (2u << 30)) // global_addr[56:32] | type=2
    };
    i32x8 g1 = {
        (int)(2u << 16),                                             // data_size=4B, mask=0
        (int)((elems & 0xFFFFu) << 16),                              // tensor_dim0[15:0]
        (int)(((elems >> 16) & 0xFFFFu) | ((rows & 0xFFFFu) << 16)), // td0[31:16] | td1[15:0]
        (int)(((rows >> 16) & 0xFFFFu) | ((elems & 0xFFFFu) << 16)), // td1[31:16] | tile_dim0
        (int)(rows & 0xFFFFu),                                       // tile_dim1 (tile_dim2=0)
        (int)row_stride,                                             // tensor_dim0_stride[31:0]
        0,                                                           // stride[47:32] | d1s[15:0]
        0
    };
#if __has_builtin(__builtin_amdgcn_tensor_load_to_lds_d2)
    __builtin_amdgcn_tensor_load_to_lds_d2(g0, g1, 0);
#else
    i32x4 gz4 = {0, 0, 0, 0};
    i32x8 gz8 = {0, 0, 0, 0, 0, 0, 0, 0};
    __builtin_amdgcn_tensor_load_to_lds(g0, g1, gz4, gz4, gz8, 0);
#endif
}

__launch_bounds__(TPB)
__global__ void voxel_query_sampler(const float* __restrict__ feat,
                                    const float* __restrict__ Kmat,
                                    const float* __restrict__ s2e,
                                    float* __restrict__ out)
{
    __shared__ float smem[GC * HW];      // 45056 B, sole LDS symbol -> offset 0

    const int b   = blockIdx.y;
    const int tid = threadIdx.x;
    const int n0  = blockIdx.x * PTS_PER_BLOCK;

    // per-batch camera params (uniform across block; broadcast loads)
    const float* M = s2e + b * 16;
    const float m00 = M[0], m01 = M[1], m02 = M[2],  tx = M[3];
    const float m10 = M[4], m11 = M[5], m12 = M[6],  ty = M[7];
    const float m20 = M[8], m21 = M[9], m22 = M[10], tz = M[11];
    const float fx = Kmat[b * 9 + 0], cx = Kmat[b * 9 + 2];
    const float fy = Kmat[b * 9 + 4], cy = Kmat[b * 9 + 5];

    // per-point gather state kept in registers for the whole channel loop
    int   i00[PPT], i10[PPT], i01[PPT], i11[PPT];
    float w00[PPT], w10[PPT], w01[PPT], w11[PPT];
    int   nn[PPT];

#pragma unroll
    for (int p = 0; p < PPT; ++p) {
        const int n = n0 + p * TPB + tid;       // consecutive across lanes
        nn[p] = n;
        const int zi = n / (VX_NX * VX_NY);
        const int r  = n - zi * (VX_NX * VX_NY);
        const int xi = r / VX_NY;
        const int yi = r - xi * VX_NY;
        const float px = xi * 0.5f - 49.75f;    // XMIN + XSTEP/2
        const float py = yi * 0.5f - 49.75f;
        const float pz = zi * 0.5f - 4.75f;     // ZMIN + ZSTEP/2
        // ego->cam: cam = R0^T * (p - t0)  (== R p + t of the reference)
        const float dx = px - tx, dy = py - ty, dz = pz - tz;
        const float cxm = m00 * dx + m10 * dy + m20 * dz;
        const float cym = m01 * dx + m11 * dy + m21 * dz;
        const float czm = m02 * dx + m12 * dy + m22 * dz;
        const float Z  = fmaxf(czm, 0.1f);
        const float u  = cxm * fx / Z + cx;
        const float v  = cym * fy / Z + cy;
        const float un = u * (2.0f / (IMG_W - 1)) - 1.0f;
        const float vn = v * (2.0f / (IMG_H - 1)) - 1.0f;
        const bool valid = (Z > 0.5f) & (un >= -1.0f) & (un <= 1.0f)
                                      & (vn >= -1.0f) & (vn <= 1.0f);
        const float ix = (un + 1.0f) * 0.5f * (IMG_W - 1);
        const float iy = (vn + 1.0f) * 0.5f * (IMG_H - 1);
        const float xf = floorf(ix), yf = floorf(iy);
        const float wx = ix - xf,   wy = iy - yf;
        const int x0 = (int)xf, y0 = (int)yf;

        auto corner = [&](int xc, int yc, float wgt, int& idx, float& w) {
            const bool inb = (xc >= 0) & (xc < IMG_W) & (yc >= 0) & (yc < IMG_H);
            const int xcc = min(max(xc, 0), IMG_W - 1);
            const int ycc = min(max(yc, 0), IMG_H - 1);
            idx = ycc * IMG_W + xcc;
            w = (valid && inb) ? wgt : 0.0f;    // select, never mask-multiply (no NaN leak)
        };
        corner(x0,     y0,     (1.0f - wx) * (1.0f - wy), i00[p], w00[p]);
        corner(x0 + 1, y0,     wx * (1.0f - wy),          i10[p], w10[p]);
        corner(x0,     y0 + 1, (1.0f - wx) * wy,          i01[p], w01[p]);
        corner(x0 + 1, y0 + 1, wx * wy,                   i11[p], w11[p]);
    }

    const size_t outB = (size_t)b * NCH * (size_t)NPTS;

    for (int g = 0; g < NSTEP; ++g) {
        __syncthreads();                         // previous step's LDS readers done
        if (tid < 32) {                          // wave 0 drives the TDM DMA
            tdm_load_rows_to_lds(0u,
                                 feat + ((size_t)b * NCH + (size_t)g * GC) * HW,
                                 HW, GC, HW);
            __builtin_amdgcn_s_wait_tensorcnt(0);
        }
        __syncthreads();                         // staged tile visible to all 8 waves

        // Tell the optimizer the DMA may have rewritten smem: the address escapes
        // into the (empty) asm together with a full memory clobber, so the ds loads
        // below cannot be folded to undef. Register operand -> no LDS mem-operand
        // selection (which crashes ISel); emits no instructions.
        asm volatile("" : : "v"(&smem[0]) : "memory");

#pragma unroll
        for (int c = 0; c < GC; ++c) {
            const float* s = smem + c * HW;
            const size_t oc = outB + (size_t)(g * GC + c) * (size_t)NPTS;
#pragma unroll
            for (int p = 0; p < PPT; ++p) {
                const float acc = w00[p] * s[i00[p]] + w10[p] * s[i10[p]]
                                + w01[p] * s[i01[p]] + w11[p] * s[i11[p]];
                // write-once 328 MB stream: keep it out of the caches
                __builtin_nontemporal_store(acc, out + oc + (size_t)nn[p]);
            }
        }
    }
}

extern "C" void kernel_launch(void* const* d_in, const int* in_sizes, int n_in,
                              void* d_out, int out_size, void* d_ws, size_t ws_size,
                              hipStream_t stream) {
    const float* feat = (const float*)d_in[0];
    const float* K    = (const float*)d_in[1];
    const float* s2e  = (const float*)d_in[2];
    float* out        = (float*)d_out;

    const int B = in_sizes[0] / (NCH * HW);      // feat is (B,64,32,88)
    dim3 grid(NPTS / PTS_PER_BLOCK, (unsigned)B);
    voxel_query_sampler<<<grid, TPB, 0, stream>>>(feat, K, s2e, out);
}